// ParameterizedAdj_412316860736
// MI455X (gfx1250) — compile-verified
//
#include <hip/hip_runtime.h>

// COO SpMM: out[row[e], :] += values[e] * x[col[e], :]  (N=50000, F=256, nnz=1.6M, fp32)
//
// MI455X reasoning: x (51.2MB) + out (51.2MB) both fit in the 192MB L2, so the
// gather/scatter is L2-bound, not HBM-bound, and WMMA has no dense structure to
// exploit. We use the CDNA5 Tensor Data Mover to stream the edge arrays
// (values/row/col) into LDS asynchronously (double-buffered, TENSORcnt-tracked),
// keeping the VMEM pipe free for x-row gathers (global_load_b128) and out-row
// scatters (global_atomic_add_f32, non-returning). TDM OOB reads return zero,
// which pads the tail chunk's values with 0.0 -> zero contribution (skipped).

#define D_FEAT  256
#define CH      1024          // edges per chunk staged via TDM
#define NWG     512           // workgroups for the main kernel
#define TPB     256           // 8 wave32s per workgroup

typedef __attribute__((ext_vector_type(4))) unsigned int u32x4;
typedef __attribute__((ext_vector_type(8))) int          i32x8;
typedef __attribute__((ext_vector_type(4))) int          i32x4;

// Issue one TDM 1-D (degenerate 2-D) tile load: CH 4-byte elements from global
// -> LDS byte offset lds_off. tensor_dim0 = elems_remaining so reads past the
// end of the array are zero-filled by the TDM (CDNA5 ISA).
__device__ __forceinline__ void tdm_load_1d(const void* gsrc, unsigned lds_off,
                                            unsigned elems_remaining) {
  unsigned long long ga = (unsigned long long)(uintptr_t)gsrc;

  u32x4 g0;
  g0[0] = 1u;                                   // count=1, is_restore=0, gather_mode=0
  g0[1] = lds_off;                              // lds_addr (bytes, wg-relative)
  g0[2] = (unsigned)(ga & 0xFFFFFFFFu);         // global_addr[31:0]   (bits 95:64)
  g0[3] = (unsigned)((ga >> 32) & 0x01FFFFFFu)  // global_addr[56:32]  (bits 120:96)
        | (2u << 30);                           // type = 2 ("image")  (bits 127:126)

  const unsigned td0   = elems_remaining;       // tensor_dim0 (elements)
  const unsigned td1   = 1u;                    // tensor_dim1
  const unsigned tile0 = CH;                    // tile_dim0 (16-bit)
  const unsigned long long s0 = CH;             // tensor_dim0_stride

  i32x8 g1;
  g1[0] = (int)(2u << 16);                                      // mask=0, data_size=2 (4B)
  g1[1] = (int)((td0 & 0xFFFFu) << 16);                         // tensor_dim0[15:0] @ 63:48
  g1[2] = (int)((td0 >> 16) | ((td1 & 0xFFFFu) << 16));         // td0[31:16] | td1[15:0]
  g1[3] = (int)((td1 >> 16) | (tile0 << 16));                   // td1[31:16] | tile_dim0
  g1[4] = 0;                                                    // tile_dim1=0, tile_dim2=0
  g1[5] = (int)(s0 & 0xFFFFFFFFull);                            // stride0[31:0]
  g1[6] = (int)((s0 >> 32) & 0xFFFFu);                          // stride0[47:32], stride1=0
  g1[7] = 0;

  i32x4 z4 = {0, 0, 0, 0};                                      // groups 2/3 unused (2-D)
  i32x8 z8 = {0, 0, 0, 0, 0, 0, 0, 0};                          // trailing group (clang-23)
  __builtin_amdgcn_tensor_load_to_lds(g0, g1, z4, z4, z8, 0);
}

__global__ void __launch_bounds__(256) zero_kernel(float4* __restrict__ out, int n4) {
  int i = blockIdx.x * blockDim.x + threadIdx.x;
  const int stride = gridDim.x * blockDim.x;
  const float4 z = make_float4(0.f, 0.f, 0.f, 0.f);
  for (; i < n4; i += stride) out[i] = z;
}

__global__ void __launch_bounds__(TPB) spmm_kernel(
    const float* __restrict__ x, const float* __restrict__ vals,
    const int* __restrict__ row, const int* __restrict__ col,
    float* __restrict__ out, int nnz) {
  // double-buffered edge staging: per buffer: vals[CH] | rows[CH] | cols[CH]
  __shared__ __align__(16) unsigned char lds[2 * CH * 12];

  const int wg   = blockIdx.x;
  const int wave = threadIdx.x >> 5;
  const int lane = threadIdx.x & 31;
  const int total_chunks = (nnz + CH - 1) / CH;

  const unsigned lds_base = (unsigned)(uintptr_t)(&lds[0]);

  auto issue = [&](int chunk, int buf) {
    const long long s = (long long)chunk * CH;
    const unsigned rem = (unsigned)(nnz - s);
    const unsigned base = lds_base + (unsigned)(buf * (CH * 12));
    tdm_load_1d(vals + s, base,          rem);
    tdm_load_1d(row  + s, base + CH * 4, rem);
    tdm_load_1d(col  + s, base + CH * 8, rem);
  };

  if (wave == 0 && wg < total_chunks) issue(wg, 0);

  int buf = 0;
  for (int c = wg; c < total_chunks; c += NWG) {
    const int cn = c + NWG;
    if (wave == 0) {
      if (cn < total_chunks) {
        issue(cn, buf ^ 1);                       // prefetch next chunk (3 TDM ops)
        __builtin_amdgcn_s_wait_tensorcnt(3);     // wait for *current* chunk's 3 ops
      } else {
        __builtin_amdgcn_s_wait_tensorcnt(0);
      }
    }
    __syncthreads();

    const unsigned char* b = lds + buf * (CH * 12);
    const float* vb = (const float*)(b);
    const int*   rb = (const int*)(b + CH * 4);
    const int*   cb = (const int*)(b + CH * 8);

    // one wave per edge; 32 lanes x 8 floats = 256 features
    for (int e = wave; e < CH; e += TPB / 32) {
      const float v = vb[e];
      if (v == 0.0f) continue;                    // TDM zero-fill tail -> no-op
      const int r  = rb[e];
      const int cc = cb[e];

      const float4* xr = (const float4*)(x + (size_t)cc * D_FEAT);
      const float4 a = xr[2 * lane];
      const float4 bb = xr[2 * lane + 1];

      float* orow = out + (size_t)r * D_FEAT + lane * 8;
      atomicAdd(orow + 0, v * a.x);
      atomicAdd(orow + 1, v * a.y);
      atomicAdd(orow + 2, v * a.z);
      atomicAdd(orow + 3, v * a.w);
      atomicAdd(orow + 4, v * bb.x);
      atomicAdd(orow + 5, v * bb.y);
      atomicAdd(orow + 6, v * bb.z);
      atomicAdd(orow + 7, v * bb.w);
    }
    __syncthreads();
    buf ^= 1;
  }
}

extern "C" void kernel_launch(void* const* d_in, const int* in_sizes, int n_in,
                              void* d_out, int out_size, void* d_ws, size_t ws_size,
                              hipStream_t stream) {
  const float* x    = (const float*)d_in[0];
  const float* vals = (const float*)d_in[1];
  const int*   row  = (const int*)d_in[2];
  const int*   col  = (const int*)d_in[3];
  float* out = (float*)d_out;
  const int nnz = in_sizes[1];

  const int n4 = out_size / 4;                    // 50000*256/4 float4s
  zero_kernel<<<2048, 256, 0, stream>>>((float4*)out, n4);
  spmm_kernel<<<NWG, TPB, 0, stream>>>(x, vals, row, col, out, nnz);
}